// DetectTM_76012331205180
// MI455X (gfx1250) — compile-verified
//
#include <hip/hip_runtime.h>
#include <stdint.h>

#define NT      360000
#define NTR     96          // 2*3*16 traces
#define MED_K   6000
#define MED_S   3000
#define NW      121         // windows per trace
#define NBINS   256
#define KRANK   2999        // (MED_K-1)/2 : lower median rank
#define RATIO   10.0f
#define POOL_H  50          // MAXPOOL_K/2
#define TOPK    100
#define MAXCAND 2048
#define TS      16384       // detection tile (samples per block)

// ---- monotonic float<->uint key transform (total order) ----
__device__ __forceinline__ unsigned ordkey(float f) {
  unsigned u = __float_as_uint(f);
  return (u & 0x80000000u) ? ~u : (u ^ 0x80000000u);
}
__device__ __forceinline__ float keyord(unsigned k) {
  unsigned u = (k & 0x80000000u) ? (k ^ 0x80000000u) : ~k;
  return __uint_as_float(u);
}

// =====================================================================
// Kernel 1: exact median + MAD per 6000-sample window via radix-select.
// One workgroup per (trace, window). Window staged into LDS with CDNA5
// async global->LDS DMA (ASYNCcnt), histograms via LDS ds_add atomics.
// =====================================================================
__global__ __launch_bounds__(256)
void mad_kernel(const float* __restrict__ x, float* __restrict__ mad_out) {
  __shared__ __align__(16) float win[MED_K];   // 24 KB
  __shared__ unsigned hist[NBINS];
  __shared__ unsigned s_pref;
  __shared__ int s_rank;

  const int tid = threadIdx.x;
  const int blk = blockIdx.x;
  const int trace = blk / NW;
  const int w = blk - trace * NW;
  const float* xt = x + (size_t)trace * NT;
  const int p0 = w * MED_S;

  const unsigned ldsbase = (unsigned)(size_t)win;

  if (p0 + MED_K <= NT) {
    // contiguous window: 16B async DMA transfers into LDS
    for (int i = tid; i < MED_K / 4; i += 256) {
      unsigned lds = ldsbase + (unsigned)i * 16u;
      unsigned long long ga = (unsigned long long)(uintptr_t)(xt + p0 + i * 4);
      asm volatile("global_load_async_to_lds_b128 %0, %1, off"
                   :: "v"(lds), "v"(ga) : "memory");
    }
  } else {
    // tail windows touch the reflect-padded region: per-element gather DMA
    for (int i = tid; i < MED_K; i += 256) {
      int p = p0 + i;
      int src = (p < NT) ? p : (2 * NT - 2 - p);   // jnp reflect pad
      unsigned lds = ldsbase + (unsigned)i * 4u;
      unsigned long long ga = (unsigned long long)(uintptr_t)(xt + src);
      asm volatile("global_load_async_to_lds_b32 %0, %1, off"
                   :: "v"(lds), "v"(ga) : "memory");
    }
  }
  asm volatile("s_wait_asynccnt 0x0" ::: "memory");
  __syncthreads();

  float med = 0.0f;
  unsigned prefix = 0u;

  for (int mode = 0; mode < 2; ++mode) {   // mode 0: median, mode 1: MAD
    prefix = 0u;
    int rank = KRANK;
    for (int b = 3; b >= 0; --b) {
      const int sh = 8 * b;
      const unsigned pm = (b == 3) ? 0u : ~((1u << (8 * (b + 1))) - 1u);
      hist[tid] = 0u;
      __syncthreads();
      for (int i = tid; i < MED_K; i += 256) {
        float v = win[i];
        if (mode) v = fabsf(v - med);
        unsigned key = ordkey(v);
        if ((key & pm) == prefix)
          atomicAdd(&hist[(key >> sh) & 0xFFu], 1u);   // ds_add
      }
      __syncthreads();
      if (tid == 0) {
        int r = rank;
        unsigned bin = 0u;
        for (int j = 0; j < NBINS; ++j) {
          int h = (int)hist[j];
          if (r < h) { bin = (unsigned)j; break; }
          r -= h;
        }
        s_pref = prefix | (bin << sh);
        s_rank = r;
      }
      __syncthreads();
      prefix = s_pref;
      rank = s_rank;
      __syncthreads();
    }
    if (mode == 0) med = keyord(prefix);
  }
  if (tid == 0) mad_out[trace * NW + w] = keyord(prefix);
}

// =====================================================================
// Kernel 2: streaming threshold + sparse local-max check + compaction.
// keep = (x is max of its clipped 101-window) && (x > 10 * mad_t).
// float4 (b128) streaming loads: 16B per vmem op to feed 23.3 TB/s.
// Since 10*MAD ~ 6.7 sigma, the 101-sample window scan almost never runs.
// =====================================================================
__device__ __forceinline__ void detect_one(const float* __restrict__ xt,
                                           const float* __restrict__ smad,
                                           int trace, int t, float v,
                                           int* __restrict__ cnt,
                                           float* __restrict__ candV,
                                           int* __restrict__ candI) {
  // bilinear upsample of MAD (align_corners=False, scale = MED_S)
  float pos = ((float)t + 0.5f) / (float)MED_S - 0.5f;
  pos = fmaxf(pos, 0.0f);
  int i0 = (int)floorf(pos);
  if (i0 > NW - 1) i0 = NW - 1;
  int i1 = i0 + 1; if (i1 > NW - 1) i1 = NW - 1;
  float wt = pos - (float)i0;
  float madt = __fadd_rn(__fmul_rn(smad[i0], 1.0f - wt),
                         __fmul_rn(smad[i1], wt));
  if (v > RATIO * madt) {
    bool ismax = true;
    int lo = t - POOL_H; if (lo < 0) lo = 0;
    int hi = t + POOL_H; if (hi > NT - 1) hi = NT - 1;
    for (int j = lo; j <= hi; ++j) {
      if (xt[j] > v) { ismax = false; break; }   // ties kept (==)
    }
    if (ismax) {
      int slot = atomicAdd(&cnt[trace], 1);
      if (slot < MAXCAND) {
        candV[trace * MAXCAND + slot] = v;
        candI[trace * MAXCAND + slot] = t;
      }
    }
  }
}

__global__ __launch_bounds__(256)
void detect_kernel(const float* __restrict__ x, const float* __restrict__ mad,
                   int* __restrict__ cnt, float* __restrict__ candV,
                   int* __restrict__ candI) {
  __shared__ float smad[NW];
  const int tid = threadIdx.x;
  const int trace = blockIdx.x;
  const int t0 = blockIdx.y * TS;
  if (tid < NW) smad[tid] = mad[trace * NW + tid];
  __syncthreads();

  const float* xt = x + (size_t)trace * NT;

  // warm the tile into cache (global_prefetch_b8)
  {
    int pf = t0 + tid * (TS / 256);
    if (pf < NT) __builtin_prefetch(xt + pf, 0, 3);
  }

  // NT % 4 == 0 and tiles are 16B aligned -> every float4 is fully in
  // range or fully out of range; no ragged tail handling needed.
  for (int i = tid * 4; i < TS; i += 256 * 4) {
    int t = t0 + i;
    if (t >= NT) break;
    float4 v4 = *reinterpret_cast<const float4*>(xt + t);
    detect_one(xt, smad, trace, t + 0, v4.x, cnt, candV, candI);
    detect_one(xt, smad, trace, t + 1, v4.y, cnt, candV, candI);
    detect_one(xt, smad, trace, t + 2, v4.z, cnt, candV, candI);
    detect_one(xt, smad, trace, t + 3, v4.w, cnt, candV, candI);
  }
}

// =====================================================================
// Kernel 3: per-trace top-100 over compacted candidates.
// 64-bit key = (ordkey(value) << 32) | ~index  -> max-key = largest value,
// ties broken toward the lower index (jax.lax.top_k semantics).
// Remaining slots: value 0 with the smallest non-candidate indices.
// =====================================================================
__global__ __launch_bounds__(256)
void topk_kernel(const float* __restrict__ candV, const int* __restrict__ candI,
                 const int* __restrict__ cnt, float* __restrict__ outV,
                 float* __restrict__ outI) {
  __shared__ unsigned long long keys[MAXCAND];   // 16 KB
  __shared__ int cidx[MAXCAND];                  // 8 KB
  __shared__ unsigned long long rk[256];
  __shared__ int rs[256];

  const int tid = threadIdx.x;
  const int trace = blockIdx.x;
  int c = cnt[trace];
  if (c > MAXCAND) c = MAXCAND;

  for (int i = tid; i < c; i += 256) {
    float v = candV[trace * MAXCAND + i];
    int id = candI[trace * MAXCAND + i];
    keys[i] = ((unsigned long long)ordkey(v) << 32) | (unsigned)(~(unsigned)id);
    cidx[i] = id;
  }
  __syncthreads();

  const int P = (c < TOPK) ? c : TOPK;
  for (int s = 0; s < P; ++s) {
    unsigned long long bk = 0ull; int bs = -1;
    for (int i = tid; i < c; i += 256)
      if (keys[i] > bk) { bk = keys[i]; bs = i; }
    rk[tid] = bk; rs[tid] = bs;
    __syncthreads();
    for (int off = 128; off > 0; off >>= 1) {
      if (tid < off && rk[tid + off] > rk[tid]) {
        rk[tid] = rk[tid + off]; rs[tid] = rs[tid + off];
      }
      __syncthreads();
    }
    if (tid == 0) {
      unsigned long long k = rk[0];
      outV[trace * TOPK + s] = keyord((unsigned)(k >> 32));
      outI[trace * TOPK + s] = (float)(int)(~(unsigned)(k & 0xFFFFFFFFull));
      keys[rs[0]] = 0ull;    // remove selected
    }
    __syncthreads();
  }

  if (tid == 0) {
    int j = 0;
    for (int s = P; s < TOPK; ++s) {
      for (;;) {
        bool isc = false;
        for (int q = 0; q < c; ++q)
          if (cidx[q] == j) { isc = true; break; }
        if (!isc) break;
        ++j;
      }
      outV[trace * TOPK + s] = 0.0f;
      outI[trace * TOPK + s] = (float)j;
      ++j;
    }
  }
}

// =====================================================================
extern "C" void kernel_launch(void* const* d_in, const int* in_sizes, int n_in,
                              void* d_out, int out_size, void* d_ws, size_t ws_size,
                              hipStream_t stream) {
  (void)in_sizes; (void)n_in; (void)out_size; (void)ws_size;
  const float* x = (const float*)d_in[0];
  float* out = (float*)d_out;            // [9600 scores][9600 indices]
  char* ws = (char*)d_ws;

  float* mad   = (float*)ws;                                    // 96*121 f32
  int*   cnt   = (int*)(ws + 64 * 1024);                        // 96 ints
  float* candV = (float*)(ws + 64 * 1024 + 4096);               // 96*2048 f32
  int*   candI = (int*)(ws + 64 * 1024 + 4096 +
                        (size_t)NTR * MAXCAND * sizeof(float)); // 96*2048 i32

  hipMemsetAsync(cnt, 0, NTR * sizeof(int), stream);
  mad_kernel<<<dim3(NTR * NW), dim3(256), 0, stream>>>(x, mad);
  detect_kernel<<<dim3(NTR, (NT + TS - 1) / TS), dim3(256), 0, stream>>>(
      x, mad, cnt, candV, candI);
  topk_kernel<<<dim3(NTR), dim3(256), 0, stream>>>(candV, candI, cnt,
                                                   out, out + NTR * TOPK);
}